// SpiralConvConvBlock_85057532330637
// MI455X (gfx1250) — compile-verified
//
#include <hip/hip_runtime.h>
#include <stdint.h>
#include <stddef.h>

#define BATCH 4
#define SEQ   4096
#define DIM   1024
#define DTILE 256     // dims per block
#define TROWS 16      // seq rows per TDM tile

#if defined(__HIP_DEVICE_COMPILE__) && defined(__gfx1250__) && __has_builtin(__builtin_amdgcn_tensor_load_to_lds)
#define HAVE_TDM 1
#else
#define HAVE_TDM 0
#endif

#if HAVE_TDM
typedef __attribute__((ext_vector_type(4))) unsigned int u32x4;
typedef __attribute__((ext_vector_type(8))) int          i32x8;
typedef __attribute__((ext_vector_type(4))) int          i32x4;

// Issue one TDM DMA: 2D tile (TROWS rows x DTILE floats, row stride DIM floats)
// from global 'gsrc' into LDS byte offset 'lds_off'.  D# layout per CDNA5 ISA §8.
__device__ __forceinline__ void tdm_load_tile(const float* gsrc, unsigned lds_off) {
  uint64_t ga = (uint64_t)(uintptr_t)gsrc;
  u32x4 g0;
  g0.x = 1u;                                             // count=1 (valid descriptor)
  g0.y = lds_off;                                        // lds_addr [63:32]
  g0.z = (unsigned)(ga & 0xFFFFFFFFull);                 // global_addr[31:0]
  g0.w = (unsigned)((ga >> 32) & 0x01FFFFFFull)          // global_addr[56:32]
       | (2u << 30);                                     // type=2 ("image")
  i32x8 g1;
  g1[0] = (int)(2u << 16);                               // wg_mask=0, data_size=2 (4B)
  g1[1] = (int)((unsigned)DTILE << 16);                  // tensor_dim0[15:0]
  g1[2] = (int)((unsigned)TROWS << 16);                  // tensor_dim0 hi=0 | tensor_dim1[15:0]
  g1[3] = (int)((unsigned)DTILE << 16);                  // tensor_dim1 hi=0 | tile_dim0
  g1[4] = (int)(unsigned)TROWS;                          // tile_dim1 | tile_dim2=0
  g1[5] = (int)(unsigned)DIM;                            // tensor_dim0_stride[31:0] (elements)
  g1[6] = 0;                                             // stride0 hi | stride1 lo
  g1[7] = 0;                                             // stride1 hi
  i32x4 z4 = {0, 0, 0, 0};                               // groups 2/3: unused (2D tensor)
  i32x8 z8 = {0, 0, 0, 0, 0, 0, 0, 0};                   // 6-arg form: extra group
  __builtin_amdgcn_tensor_load_to_lds(g0, g1, z4, z4, z8, 0);
}
#endif

// ---------------------------------------------------------------------------
// Pass A (FINAL=false): per-chunk local recurrence from 0, emit carry.
// Pass C (FINAL=true):  recurrence seeded with scanned cross-chunk state,
//                       write interleaved complex64 output.
// Grid: (DIM/DTILE, nc, BATCH), block: DTILE threads (8 wave32s).
// ---------------------------------------------------------------------------
template <bool FINAL>
__global__ __launch_bounds__(DTILE)
void chunk_kernel(const float* __restrict__ x,
                  const float* __restrict__ ph_r, const float* __restrict__ ph_i,
                  const float* __restrict__ in_r, const float* __restrict__ in_i,
                  float2* __restrict__ carry, const float2* __restrict__ Hprev,
                  float2* __restrict__ out, int Lc) {
  __shared__ float xs[2][TROWS * DTILE];                 // 32 KB double buffer
  const int tid = threadIdx.x;
  const int dt  = blockIdx.x;
  const int j   = blockIdx.y;
  const int b   = blockIdx.z;
  const int nc  = gridDim.y;
  const int d   = dt * DTILE + tid;
  const int t0  = j * Lc;
  const int nsub = Lc / TROWS;

  // phazor = p/|p| * exp(-|p|);  init = phazor_init
  float p_r = ph_r[d], p_i = ph_i[d];
  float mag = sqrtf(p_r * p_r + p_i * p_i);
  float sc  = expf(-mag) / mag;
  float zr = p_r * sc, zi = p_i * sc;
  float ir = in_r[d], ii = in_i[d];

  float hr = 0.f, hi = 0.f;
  if (FINAL) {
    float2 H = Hprev[((size_t)b * nc + j) * DIM + d];
    hr = H.x; hi = H.y;
  }

  const float* gbase = x + ((size_t)b * SEQ + t0) * DIM + (size_t)dt * DTILE;

#if HAVE_TDM
  const bool lead = (tid < 32);                          // wave 0 drives the TDM
  const unsigned lds0 = (unsigned)(uintptr_t)&xs[0][0];
  const unsigned lds1 = (unsigned)(uintptr_t)&xs[1][0];
  if (lead) tdm_load_tile(gbase, lds0);
#endif

  for (int sub = 0; sub < nsub; ++sub) {
    float* cur = &xs[sub & 1][0];
#if HAVE_TDM
    if (lead) {
      if (sub + 1 < nsub) {
        // prefetch next tile into the other buffer, then wait for current
        tdm_load_tile(gbase + (size_t)(sub + 1) * TROWS * DIM,
                      (sub & 1) ? lds0 : lds1);
        __builtin_amdgcn_s_wait_tensorcnt(1);            // current tile done
      } else {
        __builtin_amdgcn_s_wait_tensorcnt(0);            // last tile done
      }
    }
    __syncthreads();                                     // tile visible to all waves
#else
    #pragma unroll
    for (int r = 0; r < TROWS; ++r)
      cur[r * DTILE + tid] = gbase[(size_t)(sub * TROWS + r) * DIM + tid];
    __syncthreads();
#endif

    #pragma unroll
    for (int r = 0; r < TROWS; ++r) {
      float xv = cur[r * DTILE + tid];                   // ds_load, conflict-free
      float nr = fmaf(zr, hr, fmaf(-zi, hi, ir * xv));   // h = z*h + init*x
      float ni = fmaf(zr, hi, fmaf(zi, hr, ii * xv));
      hr = nr; hi = ni;
      if (FINAL) {
        int t = t0 + sub * TROWS + r;
        out[((size_t)b * SEQ + t) * DIM + d] = make_float2(hr, hi);
      }
    }
    __syncthreads();                                     // before buffer reuse
  }

  if (!FINAL)
    carry[((size_t)b * nc + j) * DIM + d] = make_float2(hr, hi);
}

// ---------------------------------------------------------------------------
// Pass B: cross-chunk scan per channel.  H_{-1} = hidden;
// Hprev[b][j][d] = H_{j-1};  H_j = z^Lc * H_{j-1} + carry_j.
// ---------------------------------------------------------------------------
__global__ __launch_bounds__(256)
void scan_kernel(const float* __restrict__ ph_r, const float* __restrict__ ph_i,
                 const float* __restrict__ hid_r, const float* __restrict__ hid_i,
                 const float2* __restrict__ carry, float2* __restrict__ Hprev,
                 int nc, int Lc) {
  const int c = blockIdx.x * blockDim.x + threadIdx.x;   // 0 .. BATCH*DIM-1
  const int b = c / DIM;
  const int d = c % DIM;

  float p_r = ph_r[d], p_i = ph_i[d];
  float mag = sqrtf(p_r * p_r + p_i * p_i);
  float sc  = expf(-mag) / mag;
  float wr = p_r * sc, wi = p_i * sc;
  for (int e = Lc; e > 1; e >>= 1) {                     // w = z^Lc (Lc = 2^k)
    float nr = wr * wr - wi * wi;
    float ni = 2.f * wr * wi;
    wr = nr; wi = ni;
  }

  float Hr = hid_r[c], Hi = hid_i[c];
  for (int j = 0; j < nc; ++j) {
    size_t idx = ((size_t)b * nc + j) * DIM + d;
    Hprev[idx] = make_float2(Hr, Hi);
    float2 cj = carry[idx];
    float nr = fmaf(wr, Hr, fmaf(-wi, Hi, cj.x));
    float ni = fmaf(wr, Hi, fmaf(wi, Hr, cj.y));
    Hr = nr; Hi = ni;
  }
}

extern "C" void kernel_launch(void* const* d_in, const int* in_sizes, int n_in,
                              void* d_out, int out_size, void* d_ws, size_t ws_size,
                              hipStream_t stream) {
  (void)in_sizes; (void)n_in; (void)out_size;
  const float* x    = (const float*)d_in[0];
  const float* hr   = (const float*)d_in[1];
  const float* hi   = (const float*)d_in[2];
  const float* pr   = (const float*)d_in[3];
  const float* pi   = (const float*)d_in[4];
  const float* qr   = (const float*)d_in[5];   // phazor_init_r
  const float* qi   = (const float*)d_in[6];   // phazor_init_i

  // choose chunk count to fit workspace (2 float2 arrays of BATCH*nc*DIM)
  int nc = 1;
  for (int cand = 32; cand >= 1; cand >>= 1) {
    size_t need = 2ull * BATCH * (size_t)cand * DIM * sizeof(float2);
    if (need <= ws_size) { nc = cand; break; }
  }
  const int Lc = SEQ / nc;

  float2* carry = (float2*)d_ws;
  float2* Hprev = carry + (size_t)BATCH * nc * DIM;
  float2* out   = (float2*)d_out;               // complex64: interleaved (re, im)

  dim3 grid(DIM / DTILE, nc, BATCH), block(DTILE);
  chunk_kernel<false><<<grid, block, 0, stream>>>(x, pr, pi, qr, qi,
                                                  carry, nullptr, nullptr, Lc);
  scan_kernel<<<(BATCH * DIM) / 256, 256, 0, stream>>>(pr, pi, hr, hi,
                                                       carry, Hprev, nc, Lc);
  chunk_kernel<true><<<grid, block, 0, stream>>>(x, pr, pi, qr, qi,
                                                 nullptr, Hprev, out, Lc);
}